// KnotNet_16561393893556
// MI455X (gfx1250) — compile-verified
//
#include <hip/hip_runtime.h>
#include <math.h>

// KnotNet on MI455X (gfx1250, wave32).
// Key insight: each braid step is a Givens rotation on the 4 strand rows,
// identical across H. The whole length-256 scan per layer == one 4x4 matrix
// R_b. Pipeline: K1 accumulate R1,R2 per batch -> K2 apply + LayerNorm to
// build X[4096,256] -> K3 MLP with V_WMMA_F32_16X16X4_F32 (exact f32 GEMM),
// X tile staged into LDS via GLOBAL_LOAD_ASYNC_TO_LDS_B128 (ASYNCcnt path).

#define BN 4096
#define LN_ 256
#define HN 64

typedef float v2f __attribute__((ext_vector_type(2)));
typedef float v8f __attribute__((ext_vector_type(8)));

// CDNA5 async global->LDS copy, 16B per lane, tracked by ASYNCcnt.
__device__ __forceinline__ void async_load_b128(unsigned lds_off, const void* gaddr) {
    asm volatile("global_load_async_to_lds_b128 %0, %1, off"
                 :: "v"(lds_off), "v"(gaddr)
                 : "memory");
}
__device__ __forceinline__ void wait_asynccnt0() {
    asm volatile("s_wait_asynccnt 0x0" ::: "memory");
}

// ---------------------------------------------------------------------------
// Kernel 1: per-batch product of 256 Givens rotations -> R1,R2 (4x4 each).
// One thread per batch; braid tiles staged through LDS (coalesced int4 loads).
// gen in {-3..3} -> 7 precomputed (cos,sin) pairs per layer in LDS.
// ---------------------------------------------------------------------------
__global__ __launch_bounds__(256) void knot_scan(const int* __restrict__ braids,
                                                 const float* __restrict__ thetas,
                                                 float* __restrict__ wsR) {
    __shared__ float2 tab[14];        // [layer][g+3] -> (cos, sin)
    __shared__ int gt[256 * 33];      // 256 batches x 32 gens, stride 33 (bank-safe)
    const int t = threadIdx.x;

    if (t < 14) {
        int layer = t / 7, gi = t % 7, g = gi - 3;
        float c = 1.0f, s = 0.0f;
        if (g != 0) {
            int a = (g < 0 ? -g : g) - 1;                    // 0..2
            float th = thetas[layer * 3 + a] * (g > 0 ? 1.0f : -1.0f);
            c = cosf(th); s = sinf(th);
        }
        tab[layer * 7 + gi] = make_float2(c, s);
    }

    const int bBase = blockIdx.x * 256;
    float r1[4][4], r2[4][4];
#pragma unroll
    for (int i = 0; i < 4; ++i)
#pragma unroll
        for (int j = 0; j < 4; ++j) {
            r1[i][j] = (i == j) ? 1.0f : 0.0f;
            r2[i][j] = (i == j) ? 1.0f : 0.0f;
        }

    for (int c = 0; c < 8; ++c) {                 // 8 chunks of 32 generators
        // cooperative, coalesced load of [256 batches][32 gens] into LDS
#pragma unroll
        for (int i = 0; i < 8; ++i) {
            int lin4 = i * 256 + t;               // int4 units, 2048 total
            int bl = lin4 >> 3;                   // 8 int4 per batch row
            int l4 = (lin4 & 7) * 4;
            const int4 g4 = *(const int4*)&braids[(size_t)(bBase + bl) * LN_ + c * 32 + l4];
            int base = bl * 33 + l4;
            gt[base + 0] = g4.x; gt[base + 1] = g4.y;
            gt[base + 2] = g4.z; gt[base + 3] = g4.w;
        }
        __syncthreads();                          // also covers tab on c==0

#pragma unroll
        for (int l = 0; l < 32; ++l) {
            int g = gt[t * 33 + l];
            int gi = g + 3;
            float2 cs1 = tab[gi];
            float2 cs2 = tab[7 + gi];
            int p = (g < 0 ? -g : g) - 1;         // -1..2 (-1 means no-op)
            bool act = (g != 0);
#pragma unroll
            for (int k = 0; k < 3; ++k) {         // branchless pair select
                bool sel = act && (p == k);
                float c1 = sel ? cs1.x : 1.0f, s1 = sel ? cs1.y : 0.0f;
                float c2 = sel ? cs2.x : 1.0f, s2 = sel ? cs2.y : 0.0f;
#pragma unroll
                for (int j = 0; j < 4; ++j) {
                    float u = r1[k][j], v = r1[k + 1][j];
                    r1[k][j]     = fmaf(u, c1, -v * s1);
                    r1[k + 1][j] = fmaf(u, s1,  v * c1);
                    float u2 = r2[k][j], v2 = r2[k + 1][j];
                    r2[k][j]     = fmaf(u2, c2, -v2 * s2);
                    r2[k + 1][j] = fmaf(u2, s2,  v2 * c2);
                }
            }
        }
        __syncthreads();
    }

    float* out = wsR + (size_t)(bBase + t) * 32;
#pragma unroll
    for (int k = 0; k < 4; ++k) {
        *(float4*)(out + k * 4)      = make_float4(r1[k][0], r1[k][1], r1[k][2], r1[k][3]);
        *(float4*)(out + 16 + k * 4) = make_float4(r2[k][0], r2[k][1], r2[k][2], r2[k][3]);
    }
}

// ---------------------------------------------------------------------------
// Kernel 2: X[b] = LN1( R2 @ LN0( R1 @ S0 ) ).  One wave per batch,
// 2 columns per lane, LayerNorm via shfl_xor butterfly (wave32).
// ---------------------------------------------------------------------------
__global__ __launch_bounds__(256) void knot_apply(const float* __restrict__ wsR,
                                                  const float* __restrict__ init,
                                                  const float* __restrict__ lnw,
                                                  const float* __restrict__ lnb,
                                                  float* __restrict__ wsX) {
    const int lane = threadIdx.x & 31;
    const int b = blockIdx.x * 8 + (threadIdx.x >> 5);
    const int j0 = lane, j1 = lane + 32;

    const float* R = wsR + (size_t)b * 32;
    float4 r1[4], r2[4];
#pragma unroll
    for (int k = 0; k < 4; ++k) {
        r1[k] = *(const float4*)(R + k * 4);
        r2[k] = *(const float4*)(R + 16 + k * 4);
    }

    float sa[4], sb[4];
#pragma unroll
    for (int k = 0; k < 4; ++k) {
        sa[k] = init[k * HN + j0];
        sb[k] = init[k * HN + j1];
    }

    float ya[4], yb[4];
#pragma unroll
    for (int layer = 0; layer < 2; ++layer) {
        const float4* r = (layer == 0) ? r1 : r2;
        float w0 = lnw[layer * HN + j0], w1v = lnw[layer * HN + j1];
        float c0 = lnb[layer * HN + j0], c1v = lnb[layer * HN + j1];
#pragma unroll
        for (int k = 0; k < 4; ++k) {
            ya[k] = r[k].x * sa[0] + r[k].y * sa[1] + r[k].z * sa[2] + r[k].w * sa[3];
            yb[k] = r[k].x * sb[0] + r[k].y * sb[1] + r[k].z * sb[2] + r[k].w * sb[3];
        }
#pragma unroll
        for (int k = 0; k < 4; ++k) {
            float s = ya[k] + yb[k];
            float q = ya[k] * ya[k] + yb[k] * yb[k];
#pragma unroll
            for (int m = 16; m >= 1; m >>= 1) {
                s += __shfl_xor(s, m, 32);
                q += __shfl_xor(q, m, 32);
            }
            float mu  = s * (1.0f / 64.0f);
            float var = q * (1.0f / 64.0f) - mu * mu;
            float inv = rsqrtf(var + 1e-5f);
            sa[k] = (ya[k] - mu) * inv * w0 + c0;   // feed next layer
            sb[k] = (yb[k] - mu) * inv * w1v + c1v;
        }
    }

#pragma unroll
    for (int k = 0; k < 4; ++k) {
        wsX[(size_t)b * 256 + k * HN + j0] = sa[k];
        wsX[(size_t)b * 256 + k * HN + j1] = sb[k];
    }
}

// ---------------------------------------------------------------------------
// Kernel 3: MLP head with V_WMMA_F32_16X16X4_F32 (exact f32).
// Block = 8 waves = 16 batches. Layer1: each wave owns one of 8 N-tiles
// (16x16 over K=256, 64 WMMAs). Layer2: waves 0..3 own 4 N-tiles (K=128,
// 32 WMMAs). Layer3 (64->2) + sigmoid on 32 lanes.
// X tile staged with GLOBAL_LOAD_ASYNC_TO_LDS_B128 (no VGPR bounce).
// LDS strides 260/132/68 words -> conflict-free b64 A-operand reads.
// ---------------------------------------------------------------------------
__global__ __launch_bounds__(256) void knot_mlp(const float* __restrict__ wsX,
                                                const float* __restrict__ w1,
                                                const float* __restrict__ b1,
                                                const float* __restrict__ w2,
                                                const float* __restrict__ b2,
                                                const float* __restrict__ w3,
                                                const float* __restrict__ b3,
                                                float* __restrict__ outp) {
    __shared__ float XS[16 * 260];
    __shared__ float H1S[16 * 132];
    __shared__ float H2S[16 * 68];

    const int t = threadIdx.x;
    const int m0 = blockIdx.x * 16;

    // stage X tile (16x256 f32) straight into LDS via async copy (ASYNCcnt)
#pragma unroll
    for (int i = 0; i < 4; ++i) {
        int lin = i * 256 + t;                 // float4 units, 1024 total
        int row = lin >> 6;
        int c4 = (lin & 63) * 4;
        unsigned lds_off = (unsigned)(size_t)&XS[row * 260 + c4];
        async_load_b128(lds_off, &wsX[(size_t)(m0 + row) * 256 + c4]);
    }
    wait_asynccnt0();
    __syncthreads();

    const int wv = t >> 5;
    const int lane = t & 31;
    const int col = lane & 15;                 // N (and A-row) index within tile
    const int kb = (lane >> 4) * 2;            // K sub-offset per f32 WMMA layout
    const int mrow = (lane >> 4) * 8;          // C/D row base per lane half

    // ---- Layer 1: h1[16,128] = relu(X @ W1^T + b1) ----
    {
        const int n0 = wv * 16;
        v8f acc = {0.f, 0.f, 0.f, 0.f, 0.f, 0.f, 0.f, 0.f};
        const float* arow = &XS[col * 260 + kb];
        const float* brow = &w1[(size_t)(n0 + col) * 256 + kb];
        for (int k = 0; k < 256; k += 4) {
            v2f a = *(const v2f*)(arow + k);
            v2f bv = *(const v2f*)(brow + k);
            acc = __builtin_amdgcn_wmma_f32_16x16x4_f32(false, a, false, bv,
                                                        (short)0, acc, false, false);
        }
        float bias = b1[n0 + col];
#pragma unroll
        for (int r = 0; r < 8; ++r) {
            float h = acc[r] + bias;
            h = h > 0.0f ? h : 0.0f;
            H1S[(mrow + r) * 132 + n0 + col] = h;
        }
    }
    __syncthreads();

    // ---- Layer 2: h2[16,64] = relu(h1 @ W2^T + b2) ----
    if (wv < 4) {                               // wave-uniform: EXEC stays all-1
        const int n0 = wv * 16;
        v8f acc = {0.f, 0.f, 0.f, 0.f, 0.f, 0.f, 0.f, 0.f};
        const float* arow = &H1S[col * 132 + kb];
        const float* brow = &w2[(size_t)(n0 + col) * 128 + kb];
        for (int k = 0; k < 128; k += 4) {
            v2f a = *(const v2f*)(arow + k);
            v2f bv = *(const v2f*)(brow + k);
            acc = __builtin_amdgcn_wmma_f32_16x16x4_f32(false, a, false, bv,
                                                        (short)0, acc, false, false);
        }
        float bias = b2[n0 + col];
#pragma unroll
        for (int r = 0; r < 8; ++r) {
            float h = acc[r] + bias;
            h = h > 0.0f ? h : 0.0f;
            H2S[(mrow + r) * 68 + n0 + col] = h;
        }
    }
    __syncthreads();

    // ---- Layer 3: out[16,2] = h2 @ W3^T + b3; sigmoid on column 0 ----
    if (t < 32) {
        int m = t & 15, o = t >> 4;
        const float* wr = &w3[o * 64];
        float s = 0.0f;
#pragma unroll
        for (int k = 0; k < 64; ++k) s = fmaf(H2S[m * 68 + k], wr[k], s);
        s += b3[o];
        if (o == 0) {
            s = 1.0f / (1.0f + expf(-s));
            outp[m0 + m] = s;
        } else {
            outp[BN + m0 + m] = s;
        }
    }
}

// ---------------------------------------------------------------------------
extern "C" void kernel_launch(void* const* d_in, const int* in_sizes, int n_in,
                              void* d_out, int out_size, void* d_ws, size_t ws_size,
                              hipStream_t stream) {
    const int*   braids = (const int*)d_in[0];
    const float* init   = (const float*)d_in[1];
    const float* thetas = (const float*)d_in[2];
    const float* lnw    = (const float*)d_in[3];
    const float* lnb    = (const float*)d_in[4];
    const float* w1     = (const float*)d_in[5];
    const float* b1     = (const float*)d_in[6];
    const float* w2     = (const float*)d_in[7];
    const float* b2     = (const float*)d_in[8];
    const float* w3     = (const float*)d_in[9];
    const float* b3     = (const float*)d_in[10];
    float* out = (float*)d_out;

    float* wsR = (float*)d_ws;             // 4096 * 32 floats = 512 KB
    float* wsX = wsR + (size_t)BN * 32;    // 4096 * 256 floats = 4 MB

    knot_scan <<<BN / 256, 256, 0, stream>>>(braids, thetas, wsR);
    knot_apply<<<BN / 8,   256, 0, stream>>>(wsR, init, lnw, lnb, wsX);
    knot_mlp  <<<BN / 16,  256, 0, stream>>>(wsX, w1, b1, w2, b2, w3, b3, out);
}